// CoLaLoLa_25692494364819
// MI455X (gfx1250) — compile-verified
//
#include <hip/hip_runtime.h>
#include <math.h>

// Problem constants (must match reference)
#define NB    2048
#define NOBJ  50
#define NTOT  128
#define FEATN 640     // NTOT*5
#define HIDN  200
#define NOUT  2

// LDS strides padded for bank-conflict-free A-fragment reads (64 banks)
#define CVS   132     // 132 % 64 = 4  -> 16 rows hit distinct bank pairs
#define ALS   644     // 644 % 64 = 4
#define HLS   212     // 212 % 64 = 20 (gcd 4 with 64, 16 rows distinct)
#define WCR   80      // w_combo rows padded 78 -> 80 (5 n-tiles of 16)
#define WCC   52      // w_combo cols padded 50 -> 52 (13 k-steps of 4)

typedef float v2f __attribute__((ext_vector_type(2)));
typedef float v8f __attribute__((ext_vector_type(8)));

// D = A(16x4,f32) * B(4x16,f32) + C(16x16,f32)  -- full-precision WMMA
static __device__ __forceinline__ v8f wmma_f32(v2f a, v2f b, v8f c) {
  return __builtin_amdgcn_wmma_f32_16x16x4_f32(false, a, false, b, (short)0, c,
                                               false, false);
}

// ---------------------------------------------------------------------------
// Kernel 1: CoLa + LoLa features.  1 wave / block, 16 events / block.
// feats[b, n*5 + {0..4}] = {mass, ptsq, weighted_e, weighted_d, weighted_pz}
// weighted_d uses the Gram expansion:
//   wd[n] = mass[n]*rowsum(w_dist)[n] + (mass @ w_dist^T)[n]
//           - 2 * sum_i metric[i] * cv_i[n] * (cv_i @ w_dist^T)[n]
// ---------------------------------------------------------------------------
__global__ __launch_bounds__(32)
void lola_feats_kernel(const float* __restrict__ vectors,
                       const float* __restrict__ w_combo,
                       const float* __restrict__ w_dist,
                       const float* __restrict__ w_ener,
                       const float* __restrict__ w_pid,
                       float* __restrict__ feats)
{
  __shared__ float cvL[4][16][CVS];   // cv components, [i][event][object]
  __shared__ float massL[16][CVS];
  __shared__ float rwL[NTOT];
  __shared__ float wcL[WCR][WCC];     // zero-padded w_combo (B-fragment source)

  const int lane = threadIdx.x;
  const int ln   = lane & 15;
  const int hi   = lane >> 4;         // half-wave select
  const int e0   = blockIdx.x << 4;   // first event of this block

  // ---- identity part of CoLa: cv[:, n<50, i] = v[:, n, i]; pad cols 50..51
  for (int idx = lane; idx < 16 * NOBJ; idx += 32) {
    const int r = idx / NOBJ, n = idx - r * NOBJ;
    const float4 v4 = *(const float4*)(vectors + (size_t)(e0 + r) * (NOBJ * 4) + n * 4);
    cvL[0][r][n] = v4.x; cvL[1][r][n] = v4.y; cvL[2][r][n] = v4.z; cvL[3][r][n] = v4.w;
  }
  if (lane < 16) {
#pragma unroll
    for (int i = 0; i < 4; ++i) { cvL[i][lane][50] = 0.f; cvL[i][lane][51] = 0.f; }
  }
  // ---- stage zero-padded w_combo into LDS (clamped unconditional loads)
  for (int idx = lane; idx < WCR * WCC; idx += 32) {
    const int row = idx / WCC, k = idx - row * WCC;
    const int rc = (row < 78) ? row : 77;
    const int kc = (k < NOBJ) ? k : NOBJ - 1;
    const float v = w_combo[rc * NOBJ + kc];
    wcL[row][k] = (row < 78 && k < NOBJ) ? v : 0.f;
  }
  // ---- rowsums of w_dist
  for (int n = lane; n < NTOT; n += 32) {
    float s = 0.f;
    for (int m = 0; m < NTOT; ++m) s += w_dist[n * NTOT + m];
    rwL[n] = s;
  }
  __syncthreads();

  // ---- learned combinations: cv[:, 50+nn, i] = v_i @ w_combo^T  (K=50 pad 52)
  for (int i = 0; i < 4; ++i) {
    v8f acc[5];
#pragma unroll
    for (int nt = 0; nt < 5; ++nt) acc[nt] = (v8f){};
    for (int ks = 0; ks < 13; ++ks) {
      const int k = ks * 4 + hi * 2;
      const v2f a = *(const v2f*)&cvL[i][ln][k];
#pragma unroll
      for (int nt = 0; nt < 5; ++nt) {
        const v2f b = *(const v2f*)&wcL[nt * 16 + ln][k];
        acc[nt] = wmma_f32(a, b, acc[nt]);
      }
    }
    __syncthreads();   // all A-fragment reads done before overwriting cols 50+
#pragma unroll
    for (int nt = 0; nt < 5; ++nt) {
      const int n = 50 + nt * 16 + ln;
      if (n < NTOT) {
#pragma unroll
        for (int v = 0; v < 8; ++v) cvL[i][v + 8 * hi][n] = acc[nt][v];
      }
    }
    __syncthreads();
  }

  // ---- masses and ptsq (metric = [-1,-1,-1,+1])
  for (int idx = lane; idx < 16 * NTOT; idx += 32) {
    const int r = idx >> 7, n = idx & 127;
    const float c0 = cvL[0][r][n], c1 = cvL[1][r][n];
    const float c2 = cvL[2][r][n], c3 = cvL[3][r][n];
    const float mass = c3 * c3 - c0 * c0 - c1 * c1 - c2 * c2;
    massL[r][n] = mass;
    float* fr = feats + (size_t)(e0 + r) * FEATN + n * 5;
    fr[0] = mass;
    fr[1] = c1 * c1 + c2 * c2;
  }
  __syncthreads();

  // ---- 7 fused K=128 GEMMs: weighted_e, weighted_pz, mass@Wd^T, cv_i@Wd^T
  for (int nt = 0; nt < 8; ++nt) {
    v8f ae = {}, ap = {}, am = {}, a0 = {}, a1 = {}, a2 = {}, a3 = {};
    const int n = nt * 16 + ln;
    for (int ks = 0; ks < 32; ++ks) {
      const int k = ks * 4 + hi * 2;
      const v2f f0 = *(const v2f*)&cvL[0][ln][k];
      const v2f f1 = *(const v2f*)&cvL[1][ln][k];
      const v2f f2 = *(const v2f*)&cvL[2][ln][k];
      const v2f f3 = *(const v2f*)&cvL[3][ln][k];
      const v2f fm = *(const v2f*)&massL[ln][k];
      const v2f be = *(const v2f*)(w_ener + n * NTOT + k);   // B = w^T
      const v2f bp = *(const v2f*)(w_pid  + n * NTOT + k);
      const v2f bd = *(const v2f*)(w_dist + n * NTOT + k);
      ae = wmma_f32(f0, be, ae);
      ap = wmma_f32(f3, bp, ap);
      am = wmma_f32(fm, bd, am);
      a0 = wmma_f32(f0, bd, a0);
      a1 = wmma_f32(f1, bd, a1);
      a2 = wmma_f32(f2, bd, a2);
      a3 = wmma_f32(f3, bd, a3);
    }
#pragma unroll
    for (int v = 0; v < 8; ++v) {
      const int r = v + 8 * hi;
      const float c0 = cvL[0][r][n], c1 = cvL[1][r][n];
      const float c2 = cvL[2][r][n], c3 = cvL[3][r][n];
      const float g  = -c0 * a0[v] - c1 * a1[v] - c2 * a2[v] + c3 * a3[v];
      const float wd = massL[r][n] * rwL[n] + am[v] - 2.f * g;
      float* fr = feats + (size_t)(e0 + r) * FEATN + n * 5;
      fr[2] = ae[v];
      fr[3] = wd;
      fr[4] = ap[v];
    }
  }
}

// ---------------------------------------------------------------------------
// Kernel 2: batch statistics per feature (deterministic shuffle reduction).
// sArr = gamma * rsqrt(var+eps);  tArr = beta - mean * sArr
// ---------------------------------------------------------------------------
__global__ __launch_bounds__(32)
void stats_kernel(const float* __restrict__ feats, const float* __restrict__ gamma,
                  const float* __restrict__ beta, float* __restrict__ sArr,
                  float* __restrict__ tArr)
{
  const int f = blockIdx.x;
  const int lane = threadIdx.x;
  float s = 0.f, q = 0.f;
  for (int e = lane; e < NB; e += 32) {
    const float x = feats[(size_t)e * FEATN + f];
    s += x; q += x * x;
  }
#pragma unroll
  for (int off = 16; off > 0; off >>= 1) {
    s += __shfl_down(s, off);
    q += __shfl_down(q, off);
  }
  if (lane == 0) {
    const float inv  = 1.f / (float)NB;
    const float mean = s * inv;
    const float var  = q * inv - mean * mean;   // biased variance (torch/jnp)
    const float sc   = gamma[f] * rsqrtf(var + 1e-5f);
    sArr[f] = sc;
    tArr[f] = beta[f] - mean * sc;
  }
}

// Kernel 3: fold BN shift into first-layer bias: b1'[n] = b1[n] + sum_k t[k]*W1[k,n]
__global__ __launch_bounds__(256)
void bias_kernel(const float* __restrict__ tArr, const float* __restrict__ W1,
                 const float* __restrict__ b1, float* __restrict__ b1p)
{
  __shared__ float tL[FEATN];
  for (int k = threadIdx.x; k < FEATN; k += 256) tL[k] = tArr[k];
  __syncthreads();
  const int n = threadIdx.x;
  if (n < HIDN) {
    float acc = b1[n];
    for (int k = 0; k < FEATN; ++k) acc += tL[k] * W1[k * HIDN + n];
    b1p[n] = acc;
  }
}

// Kernel 4: fold BN scale into activations: feats[b,k] *= s[k]
__global__ __launch_bounds__(256)
void scale_kernel(float* __restrict__ feats, const float* __restrict__ sArr)
{
  const int idx = blockIdx.x * 256 + threadIdx.x;
  if (idx < NB * FEATN) feats[idx] *= sArr[idx % FEATN];
}

// Kernel 5: transpose W1 so head B-fragments are contiguous float2 loads
__global__ __launch_bounds__(256)
void w1t_kernel(const float* __restrict__ W1, float* __restrict__ W1t)
{
  const int idx = blockIdx.x * 256 + threadIdx.x;
  if (idx < FEATN * HIDN) {
    const int n = idx / FEATN, k = idx - n * FEATN;
    W1t[idx] = W1[k * HIDN + n];
  }
}

// ---------------------------------------------------------------------------
// Kernel 6: MLP head. h = relu(featsS @ W1 + b1'), out = sigmoid(h @ W2 + b2)
// 1 wave / block, 16 events / block. N padded 200->208 (13 tiles), K=640.
// ---------------------------------------------------------------------------
__global__ __launch_bounds__(32)
void head_kernel(const float* __restrict__ featsS, const float* __restrict__ W1t,
                 const float* __restrict__ b1p, const float* __restrict__ W2,
                 const float* __restrict__ b2, float* __restrict__ out)
{
  __shared__ float aL[16][ALS];
  __shared__ float hL[16][HLS];
  const int lane = threadIdx.x;
  const int ln = lane & 15, hi = lane >> 4;
  const int e0 = blockIdx.x << 4;

  for (int idx = lane; idx < 16 * (FEATN / 4); idx += 32) {
    const int r = idx / (FEATN / 4), c = idx - r * (FEATN / 4);
    const float4 t = *(const float4*)(featsS + (size_t)(e0 + r) * FEATN + c * 4);
    *(float4*)&aL[r][c * 4] = t;
  }
  __syncthreads();

  for (int nt = 0; nt < 13; ++nt) {
    v8f acc = {};
    const int n = nt * 16 + ln;
    const bool nv = (n < HIDN);
    const int nc = nv ? n : 0;
    for (int ks = 0; ks < FEATN / 4; ++ks) {
      const int k = ks * 4 + hi * 2;
      const v2f a = *(const v2f*)&aL[ln][k];
      v2f b = *(const v2f*)(W1t + (size_t)nc * FEATN + k);
      b.x = nv ? b.x : 0.f;
      b.y = nv ? b.y : 0.f;
      acc = wmma_f32(a, b, acc);
    }
    if (nv) {
      const float bias = b1p[n];
#pragma unroll
      for (int v = 0; v < 8; ++v) hL[v + 8 * hi][n] = fmaxf(acc[v] + bias, 0.f);
    }
  }
  __syncthreads();

  // 16 events x 2 outputs == 32 lanes: one 200-length dot each, then sigmoid
  const int e = lane >> 1, j = lane & 1;
  float acc = b2[j];
  for (int n = 0; n < HIDN; ++n) acc += hL[e][n] * W2[n * NOUT + j];
  out[(size_t)(e0 + e) * NOUT + j] = 1.f / (1.f + expf(-acc));
}

// ---------------------------------------------------------------------------
extern "C" void kernel_launch(void* const* d_in, const int* in_sizes, int n_in,
                              void* d_out, int out_size, void* d_ws, size_t ws_size,
                              hipStream_t stream)
{
  (void)in_sizes; (void)n_in; (void)out_size; (void)ws_size;
  const float* vectors = (const float*)d_in[0];
  const float* w_combo = (const float*)d_in[1];
  const float* w_dist  = (const float*)d_in[2];
  const float* w_ener  = (const float*)d_in[3];
  const float* w_pid   = (const float*)d_in[4];
  const float* gamma   = (const float*)d_in[5];
  const float* beta    = (const float*)d_in[6];
  const float* W1      = (const float*)d_in[7];
  const float* b1      = (const float*)d_in[8];
  const float* W2      = (const float*)d_in[9];
  const float* b2      = (const float*)d_in[10];
  float* out = (float*)d_out;

  // workspace layout (floats): feats | sArr | tArr | b1p(pad) | W1t
  float* ws    = (float*)d_ws;
  float* feats = ws;                               // NB*FEATN   = 1,310,720
  float* sArr  = feats + (size_t)NB * FEATN;       // 640
  float* tArr  = sArr + FEATN;                     // 640
  float* b1p   = tArr + FEATN;                     // 200 (pad 256)
  float* W1t   = b1p + 256;                        // 640*200 = 128,000

  w1t_kernel<<<(FEATN * HIDN + 255) / 256, 256, 0, stream>>>(W1, W1t);
  lola_feats_kernel<<<NB / 16, 32, 0, stream>>>(vectors, w_combo, w_dist,
                                                w_ener, w_pid, feats);
  stats_kernel<<<FEATN, 32, 0, stream>>>(feats, gamma, beta, sArr, tArr);
  bias_kernel<<<1, 256, 0, stream>>>(tArr, W1, b1, b1p);
  scale_kernel<<<(NB * FEATN) / 256, 256, 0, stream>>>(feats, sArr);
  head_kernel<<<NB / 16, 32, 0, stream>>>(feats, W1t, b1p, W2, b2, out);
}